// MaeConsistentContrastiveLoss_42786464202874
// MI455X (gfx1250) — compile-verified
//
#include <hip/hip_runtime.h>
#include <math.h>

typedef __attribute__((ext_vector_type(2))) float v2f;
typedef __attribute__((ext_vector_type(8))) float v8f;

#define Bn   48
#define Ln   17
#define Dn   256
#define Mtok 768   // Bn*(Ln-1)

// ---------------------------------------------------------------------------
// Kernel 1: L2-normalize rows of x (48,17,256); scatter into anchors / toks.
// One wave (32 lanes) per row, 8 floats per lane. Also zero the output scalar.
// ---------------------------------------------------------------------------
__global__ __launch_bounds__(256) void normalize_kernel(
    const float* __restrict__ x,
    float* __restrict__ anchors,
    float* __restrict__ toks,
    float* __restrict__ out)
{
  if (blockIdx.x == 0 && threadIdx.x == 0) out[0] = 0.0f;
  int row  = blockIdx.x * 8 + (threadIdx.x >> 5);
  int lane = threadIdx.x & 31;
  if (row >= Bn * Ln) return;

  const float4* src = (const float4*)(x + (size_t)row * Dn);
  float4 a = src[lane];
  float4 b = src[lane + 32];
  float s = a.x*a.x + a.y*a.y + a.z*a.z + a.w*a.w
          + b.x*b.x + b.y*b.y + b.z*b.z + b.w*b.w;
  #pragma unroll
  for (int off = 16; off > 0; off >>= 1) s += __shfl_xor(s, off, 32);
  float inv = 1.0f / fmaxf(sqrtf(s), 1e-12f);

  int bb = row / Ln, l = row % Ln;
  float* dstp = (l == 0) ? (anchors + (size_t)bb * Dn)
                         : (toks + (size_t)(bb * (Ln - 1) + (l - 1)) * Dn);
  a.x *= inv; a.y *= inv; a.z *= inv; a.w *= inv;
  b.x *= inv; b.y *= inv; b.z *= inv; b.w *= inv;
  float4* dst = (float4*)dstp;
  dst[lane]      = a;
  dst[lane + 32] = b;
}

// ---------------------------------------------------------------------------
// Kernel 2: C[i,j] = sum_k A[i,k]*B[j,k]  (K = 256, row-major, B transposed)
// One wave per 16x16 tile, V_WMMA_F32_16X16X4_F32, exact fp32.
// f32 A fragment: lane<16 holds (K0,K1), lane>=16 holds (K2,K3) for row=t&15;
// the B fragment mirrors this -> both operands are a single float2 load.
// ---------------------------------------------------------------------------
__global__ __launch_bounds__(32) void wmma_gemm_nt(
    const float* __restrict__ A,
    const float* __restrict__ Bm,
    float* __restrict__ C,
    int N)
{
  int i0 = blockIdx.y * 16;
  int j0 = blockIdx.x * 16;
  int t  = threadIdx.x;
  int half = t >> 4;      // 0 or 1
  int tl   = t & 15;

  const float* arow = A  + (size_t)(i0 + tl) * Dn + 2 * half;
  const float* brow = Bm + (size_t)(j0 + tl) * Dn + 2 * half;

  v8f acc0 = {};
  v8f acc1 = {};
  #pragma unroll 8
  for (int k = 0; k < Dn; k += 8) {
    v2f a0 = *(const v2f*)(arow + k);
    v2f b0 = *(const v2f*)(brow + k);
    acc0 = __builtin_amdgcn_wmma_f32_16x16x4_f32(false, a0, false, b0,
                                                 (short)0, acc0, false, false);
    v2f a1 = *(const v2f*)(arow + k + 4);
    v2f b1 = *(const v2f*)(brow + k + 4);
    acc1 = __builtin_amdgcn_wmma_f32_16x16x4_f32(false, a1, false, b1,
                                                 (short)0, acc1, false, false);
  }
  v8f acc = acc0 + acc1;

  int col   = j0 + tl;
  int rbase = i0 + 8 * half;   // C/D layout: VGPR v holds row rbase+v, col=lane&15
  #pragma unroll
  for (int v = 0; v < 8; ++v)
    C[(size_t)(rbase + v) * N + col] = acc[v];
}

// ---------------------------------------------------------------------------
// Block reductions (256 threads)
// ---------------------------------------------------------------------------
__device__ inline float block_reduce_sum(float v, float* red, int t) {
  __syncthreads();
  red[t] = v;
  __syncthreads();
  for (int s = 128; s > 0; s >>= 1) {
    if (t < s) red[t] += red[t + s];
    __syncthreads();
  }
  float r = red[0];
  __syncthreads();
  return r;
}
__device__ inline float block_reduce_max(float v, float* red, int t) {
  __syncthreads();
  red[t] = v;
  __syncthreads();
  for (int s = 128; s > 0; s >>= 1) {
    if (t < s) red[t] = fmaxf(red[t], red[t + s]);
    __syncthreads();
  }
  float r = red[0];
  __syncthreads();
  return r;
}

// ---------------------------------------------------------------------------
// Kernel 3: per-b masked softmax / symmetric-KL reduction.
// kl1[j] = sum_neg smp*(p-a)        + (La - Lp)          (since sum_neg smp = 1)
// kl2[j] = Qa - sum_neg sma*p + Sa*(Lp - La)
// One block per b (8 waves); wave-per-j, 24 s_tok elements per lane in regs.
// ---------------------------------------------------------------------------
__global__ __launch_bounds__(256) void loss_kernel(
    const float* __restrict__ s_anc,
    const float* __restrict__ s_tok,
    const int*  __restrict__ label,
    float* __restrict__ out)
{
  __shared__ float am[Mtok];
  __shared__ float sma[Mtok];
  __shared__ float negf[Mtok];
  __shared__ float red[256];
  __shared__ int   lab[Bn];
  __shared__ float sh_ld, sh_La, sh_Sa, sh_Qa, sh_Pb;
  __shared__ float wacc[8];

  int b = blockIdx.x;
  int t = threadIdx.x;
  if (t < Bn) lab[t] = label[t];
  __syncthreads();
  int lb = lab[b];

  // --- per-b setup: masks, log_denom ---
  float ngv[3], sav[3];
  float m0 = -3.4e38f;
  #pragma unroll
  for (int i = 0; i < 3; ++i) {
    int k = t + 256 * i;
    float ng = (lab[k >> 4] != lb) ? 1.0f : 0.0f;   // owner(k) = k/16
    ngv[i] = ng;
    float sv = s_anc[(size_t)b * Mtok + k];
    sav[i] = sv;
    negf[k] = ng;
    m0 = fmaxf(m0, (ng > 0.0f) ? sv : -3.4e38f);
  }
  m0 = block_reduce_max(m0, red, t);
  float zs = 0.0f, pc = 0.0f;
  #pragma unroll
  for (int i = 0; i < 3; ++i) {
    zs += ngv[i] * __expf(sav[i] - m0);
    pc += 1.0f - ngv[i];                            // count of "same" = P[b]
  }
  zs = block_reduce_sum(zs, red, t);
  pc = block_reduce_sum(pc, red, t);
  if (t == 0) { sh_ld = m0 + __logf(zs); sh_Pb = pc; }
  __syncthreads();
  float ld = sh_ld;

  // --- a_m = exp(s_anc - ld) on negatives; La = LSE(a_m); sm_a; Sa; Qa ---
  float amv[3];
  float m2 = -3.4e38f;
  #pragma unroll
  for (int i = 0; i < 3; ++i) {
    int k = t + 256 * i;
    float av = (ngv[i] > 0.0f) ? __expf(sav[i] - ld) : 0.0f;
    amv[i] = av;
    am[k] = av;
    m2 = fmaxf(m2, (ngv[i] > 0.0f) ? av : -3.4e38f);
  }
  m2 = block_reduce_max(m2, red, t);
  float z2 = 0.0f;
  #pragma unroll
  for (int i = 0; i < 3; ++i) z2 += ngv[i] * __expf(amv[i] - m2);
  z2 = block_reduce_sum(z2, red, t);
  float La = m2 + __logf(z2);
  float sa_p = 0.0f, qa_p = 0.0f;
  #pragma unroll
  for (int i = 0; i < 3; ++i) {
    int k = t + 256 * i;
    float sv = ngv[i] * __expf(amv[i] - La);
    sma[k] = sv;
    sa_p += sv;
    qa_p += sv * amv[i];
  }
  float Sa = block_reduce_sum(sa_p, red, t);
  float Qa = block_reduce_sum(qa_p, red, t);
  if (t == 0) { sh_La = La; sh_Sa = Sa; sh_Qa = Qa; }
  __syncthreads();
  La = sh_La; Sa = sh_Sa; Qa = sh_Qa;

  // --- wave-per-j KL accumulation over same-label rows j ---
  int w = t >> 5, lane = t & 31;
  float acc = 0.0f;
  for (int j = w; j < Mtok; j += 8) {
    if (negf[j] != 0.0f) continue;        // only rows with same[b,j]
    const float* srow = s_tok + (size_t)j * Mtok;
    float p[24];
    float m = -3.4e38f;
    #pragma unroll
    for (int i = 0; i < 24; ++i) {
      int k = lane + 32 * i;
      float pv = __expf(srow[k] - ld);
      p[i] = pv;
      m = fmaxf(m, (negf[k] > 0.0f) ? pv : -3.4e38f);
    }
    #pragma unroll
    for (int off = 16; off > 0; off >>= 1) m = fmaxf(m, __shfl_xor(m, off, 32));
    float z = 0.0f, tt = 0.0f, u = 0.0f;
    #pragma unroll
    for (int i = 0; i < 24; ++i) {
      int k = lane + 32 * i;
      float ng = negf[k];
      float e  = __expf(p[i] - m);
      z  += ng * e;
      tt += ng * e * (p[i] - am[k]);
      u  += sma[k] * p[i];
    }
    #pragma unroll
    for (int off = 16; off > 0; off >>= 1) {
      z  += __shfl_xor(z, off, 32);
      tt += __shfl_xor(tt, off, 32);
      u  += __shfl_xor(u, off, 32);
    }
    if (lane == 0) {
      float Lp  = m + __logf(z);
      float kl1 = tt / z + (La - Lp);
      float kl2 = Qa - u + Sa * (Lp - La);
      acc += 0.5f * (kl1 + kl2);
    }
  }
  if (lane == 0) wacc[w] = acc;
  __syncthreads();
  if (t == 0) {
    float s = 0.0f;
    #pragma unroll
    for (int i = 0; i < 8; ++i) s += wacc[i];
    atomicAdd(out, s / sh_Pb * (1.0f / (float)Bn));
  }
}

// ---------------------------------------------------------------------------
extern "C" void kernel_launch(void* const* d_in, const int* in_sizes, int n_in,
                              void* d_out, int out_size, void* d_ws, size_t ws_size,
                              hipStream_t stream) {
  (void)in_sizes; (void)n_in; (void)out_size; (void)ws_size;
  const float* x   = (const float*)d_in[0];
  const int*   lab = (const int*)d_in[1];
  float* out = (float*)d_out;

  float* ws      = (float*)d_ws;
  float* anchors = ws;                       // 48*256   = 12288
  float* toks    = ws + 12288;               // 768*256  = 196608
  float* s_anc   = ws + 12288 + 196608;      // 48*768   = 36864
  float* s_tok   = s_anc + 36864;            // 768*768  = 589824

  normalize_kernel<<<102, 256, 0, stream>>>(x, anchors, toks, out);
  wmma_gemm_nt<<<dim3(Mtok / 16, Bn / 16), 32, 0, stream>>>(anchors, toks, s_anc, Mtok);
  wmma_gemm_nt<<<dim3(Mtok / 16, Mtok / 16), 32, 0, stream>>>(toks, toks, s_tok, Mtok);
  loss_kernel<<<Bn, 256, 0, stream>>>(s_anc, s_tok, lab, out);
}